// SeqSelfAttention_76862734729706
// MI455X (gfx1250) — compile-verified
//
#include <hip/hip_runtime.h>
#include <cmath>

typedef __attribute__((ext_vector_type(2))) float v2f;
typedef __attribute__((ext_vector_type(8))) float v8f;

#define B_ 4
#define T_ 1024
#define D_ 512
#define P_ 64
#define M_ 256

__device__ __forceinline__ float fast_tanh(float x) {
#if __has_builtin(__builtin_amdgcn_tanhf)
  return __builtin_amdgcn_tanhf(x);
#else
  float r;
  asm("v_tanh_f32 %0, %1" : "=v"(r) : "v"(x));   // gfx1250 TRANS op
  return r;
#endif
}

// ---------------------------------------------------------------------------
// Kernel A: w1h = x@W1+b1 and w2h = x@W2+b2 fused (shared A fragment).
// One wave computes a 16x16 tile of BOTH outputs via V_WMMA_F32_16X16X4_F32.
// A 16x4 layout: lane m = lane&15, K pair = (lane>>4)*2 (per ISA 7.12.2).
// ---------------------------------------------------------------------------
__global__ void proj_kernel(const float* __restrict__ x,   // (B*T, D)
                            const float* __restrict__ W1,  // (D, P)
                            const float* __restrict__ b1,  // (P)
                            const float* __restrict__ W2,  // (D, P)
                            const float* __restrict__ b2,  // (P)
                            float* __restrict__ w1h,       // (B*T, P)
                            float* __restrict__ w2h) {     // (B*T, P)
  const int lane  = threadIdx.x & 31;
  const int wave  = threadIdx.x >> 5;          // 8 waves / block
  const int ntile = wave & 3;                  // P/16 = 4 tiles
  const int rtile = blockIdx.x * 2 + (wave >> 2);
  const int row0  = rtile * 16;
  const int n0    = ntile * 16;

  const int mi = lane & 15;                    // M (for A) / N (for B)
  const int kb = (lane >> 4) * 2;              // K sub-pair within step
  const int n  = n0 + mi;

  v8f acc1 = {};
  v8f acc2 = {};
  for (int k = 0; k < D_; k += 4) {
    v2f a, bA, bB;
    const float* ap = x + (size_t)(row0 + mi) * D_ + (k + kb);
    a.x = ap[0];
    a.y = ap[1];
    const float* w1p = W1 + (size_t)(k + kb) * P_ + n;
    bA.x = w1p[0];
    bA.y = w1p[P_];
    const float* w2p = W2 + (size_t)(k + kb) * P_ + n;
    bB.x = w2p[0];
    bB.y = w2p[P_];
    acc1 = __builtin_amdgcn_wmma_f32_16x16x4_f32(false, a, false, bA,
                                                 (short)0, acc1, false, false);
    acc2 = __builtin_amdgcn_wmma_f32_16x16x4_f32(false, a, false, bB,
                                                 (short)0, acc2, false, false);
  }

  const int m_off = (lane >> 4) * 8;           // C/D layout: M split at lane 16
  const float bias1 = b1[n];
  const float bias2 = b2[n];
#pragma unroll
  for (int i = 0; i < 8; ++i) {
    const int m = row0 + m_off + i;
    w1h[(size_t)m * P_ + n] = acc1[i] + bias1;
    w2h[(size_t)m * P_ + n] = acc2[i] + bias2;
  }
}

// ---------------------------------------------------------------------------
// Kernel B: w3e = ec@W3 + b3   (tiny: 4x64 outputs, K=256)
// ---------------------------------------------------------------------------
__global__ void ctx_kernel(const float* __restrict__ ec,   // (B, M)
                           const float* __restrict__ W3,   // (M, P)
                           const float* __restrict__ b3,   // (P)
                           float* __restrict__ w3e) {      // (B, P)
  const int b = blockIdx.x;
  const int p = threadIdx.x;
  float acc = b3[p];
  for (int m = 0; m < M_; ++m)
    acc = fmaf(ec[b * M_ + m], W3[m * P_ + p], acc);
  w3e[b * P_ + p] = acc;
}

// ---------------------------------------------------------------------------
// Kernel C: fused additive-attention scores (the tanh hot loop).
// at[b,t,u] = sum_p wa_w[p]*tanh(w1h[b,t,p] + w2h[b,u,p] + w3e[b,p]) + wa_b
// 32x32 (t,u) tile per 256-thread block; gt never touches memory.
// LDS rows padded to 66 floats -> conflict-free across lanes.
// ---------------------------------------------------------------------------
__global__ void score_kernel(const float* __restrict__ w1h,
                             const float* __restrict__ w2h,
                             const float* __restrict__ w3e,
                             const float* __restrict__ wa_w,
                             const float* __restrict__ wa_b_p,
                             float* __restrict__ at) {      // (B, T, T)
  __shared__ float s1[32][66];
  __shared__ float s2[32][66];
  __shared__ float sw[64];

  const int b  = blockIdx.z;
  const int t0 = blockIdx.y * 32;
  const int u0 = blockIdx.x * 32;
  const int tid = threadIdx.x;

  if (tid < 64) sw[tid] = wa_w[tid];
  for (int i = tid; i < 32 * 64; i += 256) {
    const int r = i >> 6;
    const int c = i & 63;
    s1[r][c] = w1h[((size_t)b * T_ + t0 + r) * P_ + c] + w3e[b * P_ + c];
    s2[r][c] = w2h[((size_t)b * T_ + u0 + r) * P_ + c];
  }
  __syncthreads();

  const float wab = wa_b_p[0];
#pragma unroll
  for (int j = 0; j < 4; ++j) {
    const int idx = tid + 256 * j;   // 0..1023
    const int ti  = idx >> 5;        // broadcast within a wave (free LDS read)
    const int uj  = idx & 31;        // = lane -> conflict-free with stride 66
    float acc = 0.0f;
#pragma unroll 8
    for (int p = 0; p < 64; ++p)
      acc = fmaf(sw[p], fast_tanh(s1[ti][p] + s2[uj][p]), acc);
    at[((size_t)b * T_ + (t0 + ti)) * T_ + (u0 + uj)] = acc + wab;
  }
}

// ---------------------------------------------------------------------------
// Kernel D: row softmax over u (row length T=1024), one block per (b,t).
// ---------------------------------------------------------------------------
__global__ void softmax_kernel(float* __restrict__ at) {
  __shared__ float red[256];
  float* r = at + (size_t)blockIdx.x * T_;
  const int tid = threadIdx.x;

  float v[4];
  float mx = -3.0e38f;
#pragma unroll
  for (int j = 0; j < 4; ++j) {
    v[j] = r[tid + 256 * j];
    mx = fmaxf(mx, v[j]);
  }
  red[tid] = mx;
  __syncthreads();
  for (int s = 128; s > 0; s >>= 1) {
    if (tid < s) red[tid] = fmaxf(red[tid], red[tid + s]);
    __syncthreads();
  }
  mx = red[0];
  __syncthreads();

  float sum = 0.0f;
#pragma unroll
  for (int j = 0; j < 4; ++j) {
    v[j] = __expf(v[j] - mx);        // v_exp_f32
    sum += v[j];
  }
  red[tid] = sum;
  __syncthreads();
  for (int s = 128; s > 0; s >>= 1) {
    if (tid < s) red[tid] += red[tid + s];
    __syncthreads();
  }
  const float inv = 1.0f / red[0];
#pragma unroll
  for (int j = 0; j < 4; ++j) r[tid + 256 * j] = v[j] * inv;
}

// ---------------------------------------------------------------------------
// Kernel E: out = at @ x  per batch: (1024x1024)@(1024x512) fp32 WMMA,
// LDS-tiled. Block tile: 32(t) x 128(d), K chunks of 32.
// Coalesced float4 global loads -> LDS; waves read WMMA fragments from LDS.
//   sA stride 34: even (b64-aligned), conflict-free across 16 m-lanes.
//   sB stride 136: 2*136 mod 64 == 16 -> kb=0 half-wave uses banks 0..15,
//   kb=2 half-wave uses banks 16..31 (disjoint).
// ---------------------------------------------------------------------------
__global__ void out_gemm_kernel(const float* __restrict__ at,  // (B, T, T)
                                const float* __restrict__ x,   // (B, T, D)
                                float* __restrict__ out) {     // (B, T, D)
  __shared__ float sA[32][34];
  __shared__ float sB[32][136];

  const int tid  = threadIdx.x;
  const int lane = tid & 31;
  const int wave = tid >> 5;

  // 512 blocks = B(4) * Mblocks(32) * Nblocks(4)
  const int b    = blockIdx.x >> 7;     // 128 blocks per batch
  const int blk  = blockIdx.x & 127;
  const int t0b  = (blk >> 2) * 32;     // M block origin
  const int d0b  = (blk & 3) * 128;     // N block origin

  const int msub = wave >> 2;           // 0..1 : which 16-row sub-tile
  const int nsub = wave & 3;            // 0..3 : which 32-col sub-tile
  const int t0   = t0b + msub * 16;
  const int dloc = nsub * 32;

  const float* A = at + (size_t)b * T_ * T_;
  const float* X = x + (size_t)b * T_ * D_;

  const int mi = lane & 15;
  const int kb = (lane >> 4) * 2;

  // cooperative-load indices
  const int arow = tid >> 3;            // 0..31
  const int acol = (tid & 7) * 4;       // 0..28
  const int brow = tid >> 5;            // 0..7 (+8 per iter)
  const int bcol = (tid & 31) * 4;      // 0..124

  v8f acc0 = {};
  v8f acc1 = {};

  for (int k0 = 0; k0 < T_; k0 += 32) {
    // A tile: 32 x 32 (coalesced b128 loads)
    {
      const float4 va =
          *(const float4*)(A + (size_t)(t0b + arow) * T_ + k0 + acol);
      sA[arow][acol + 0] = va.x;
      sA[arow][acol + 1] = va.y;
      sA[arow][acol + 2] = va.z;
      sA[arow][acol + 3] = va.w;
    }
    // B tile: 32 x 128 (coalesced b128 loads)
#pragma unroll
    for (int it = 0; it < 4; ++it) {
      const int r = brow + it * 8;
      const float4 vb =
          *(const float4*)(X + (size_t)(k0 + r) * D_ + d0b + bcol);
      sB[r][bcol + 0] = vb.x;
      sB[r][bcol + 1] = vb.y;
      sB[r][bcol + 2] = vb.z;
      sB[r][bcol + 3] = vb.w;
    }
    __syncthreads();

#pragma unroll
    for (int kk = 0; kk < 32; kk += 4) {
      v2f a, bb0, bb1;
      a.x  = sA[msub * 16 + mi][kk + kb];
      a.y  = sA[msub * 16 + mi][kk + kb + 1];
      bb0.x = sB[kk + kb][dloc + mi];
      bb0.y = sB[kk + kb + 1][dloc + mi];
      bb1.x = sB[kk + kb][dloc + 16 + mi];
      bb1.y = sB[kk + kb + 1][dloc + 16 + mi];
      acc0 = __builtin_amdgcn_wmma_f32_16x16x4_f32(false, a, false, bb0,
                                                   (short)0, acc0, false, false);
      acc1 = __builtin_amdgcn_wmma_f32_16x16x4_f32(false, a, false, bb1,
                                                   (short)0, acc1, false, false);
    }
    __syncthreads();
  }

  const int m_off = (lane >> 4) * 8;
#pragma unroll
  for (int i = 0; i < 8; ++i) {
    const int t = t0 + m_off + i;
    out[((size_t)b * T_ + t) * D_ + d0b + dloc + mi]      = acc0[i];
    out[((size_t)b * T_ + t) * D_ + d0b + dloc + 16 + mi] = acc1[i];
  }
}

// ---------------------------------------------------------------------------
extern "C" void kernel_launch(void* const* d_in, const int* in_sizes, int n_in,
                              void* d_out, int out_size, void* d_ws, size_t ws_size,
                              hipStream_t stream) {
  (void)in_sizes; (void)n_in; (void)out_size; (void)ws_size;

  const float* x    = (const float*)d_in[0];  // (B,T,D)
  const float* ec   = (const float*)d_in[1];  // (B,M)
  const float* W1   = (const float*)d_in[2];  // (D,P)
  const float* b1   = (const float*)d_in[3];  // (P)
  const float* W2   = (const float*)d_in[4];  // (D,P)
  const float* b2   = (const float*)d_in[5];  // (P)
  const float* W3   = (const float*)d_in[6];  // (M,P)
  const float* b3   = (const float*)d_in[7];  // (P)
  const float* wa_w = (const float*)d_in[8];  // (P)
  const float* wa_b = (const float*)d_in[9];  // scalar
  float* out = (float*)d_out;

  // Workspace layout (floats)
  float* ws  = (float*)d_ws;
  float* w1h = ws;                           // B*T*P = 262144
  float* w2h = w1h + (size_t)B_ * T_ * P_;   // 262144
  float* w3e = w2h + (size_t)B_ * T_ * P_;   // 256 (padded to 512)
  float* at  = w3e + 512;                    // B*T*T = 4194304

  // A: projections (fused W1/W2 WMMA GEMM). 4096 rows /16 /2-per-block = 128.
  proj_kernel<<<128, 256, 0, stream>>>(x, W1, b1, W2, b2, w1h, w2h);

  // B: context projection
  ctx_kernel<<<4, 64, 0, stream>>>(ec, W3, b3, w3e);

  // C: fused tanh additive scores
  score_kernel<<<dim3(T_ / 32, T_ / 32, B_), 256, 0, stream>>>(
      w1h, w2h, w3e, wa_w, wa_b, at);

  // D: softmax over last axis
  softmax_kernel<<<B_ * T_, 256, 0, stream>>>(at);

  // E: out = at @ x  (LDS-tiled WMMA fp32). 512 blocks.
  out_gemm_kernel<<<512, 256, 0, stream>>>(at, x, out);
}